// STARFM_70368744178201
// MI455X (gfx1250) — compile-verified
//
#include <hip/hip_runtime.h>
#include <math.h>

// ---- problem constants (from reference) ----
#define B_      2
#define C_      4
#define T_      2
#define HIMG    110
#define WIMG    110
#define WS_     15
#define OH_     96
#define TILE    16
#define REG     30          // TILE + WS - 1
#define RSTRIDE 32          // padded LDS row stride (floats)
#define EPS_F   1e-8f
#define SPEC_U  1.41421356237309515e-3f   // sqrt(2)*1e-3
#define TEMP_U  1.41421356237309515e-3f
#define PLANE   (HIMG * WIMG)

// async global->LDS path (gfx1250): guarded so the file compiles on any toolchain
#if defined(__gfx1250__) && __has_builtin(__builtin_amdgcn_global_load_async_to_lds_b32)
#define HAVE_ASYNC_LDS 1
#else
#define HAVE_ASYNC_LDS 0
#endif

typedef __attribute__((address_space(1))) int* g_i32p;   // global
typedef __attribute__((address_space(3))) int* l_i32p;   // LDS

__device__ __forceinline__ void copy_f32_to_lds(const float* g, float* l) {
#if HAVE_ASYNC_LDS
  __builtin_amdgcn_global_load_async_to_lds_b32((g_i32p)g, (l_i32p)l, 0, 0);
#else
  *l = *g;
#endif
}

__device__ __forceinline__ void wait_async_lds() {
#if HAVE_ASYNC_LDS
#if __has_builtin(__builtin_amdgcn_s_wait_asynccnt)
  __builtin_amdgcn_s_wait_asynccnt(0);
#else
  asm volatile("s_wait_asynccnt 0" ::: "memory");
#endif
#endif
}

// fast reciprocal: v_rcp_f32 (weight is renormalized later, ~1ulp is plenty)
__device__ __forceinline__ float fast_rcp(float x) {
#if __has_builtin(__builtin_amdgcn_rcpf)
  return __builtin_amdgcn_rcpf(x);
#else
  return 1.0f / x;
#endif
}

// window multiplicity of image coordinate v (how many 15-wide windows cover it)
__device__ __forceinline__ int multv(int v) {
  int lo = v - (WS_ - 1); if (lo < 0) lo = 0;
  int hi = v;             if (hi > OH_ - 1) hi = OH_ - 1;
  return hi - lo + 1;
}

// ---------------------------------------------------------------------------
// Pass 1: deterministic single-block reduction over prior_fine, weighted by
// window multiplicity, producing the global similarity threshold thr.
// ---------------------------------------------------------------------------
__global__ void __launch_bounds__(256)
starfm_stats(const float* __restrict__ pf, float* __restrict__ thr_out) {
  __shared__ double sc[256], s1[256], s2[256];
  const int tid = threadIdx.x;
  double c = 0.0, a1 = 0.0, a2 = 0.0;
  const int N = B_ * C_ * T_ * PLANE;
  for (int i = tid; i < N; i += 256) {
    const int xy = i % PLANE;
    const int y = xy / WIMG, x = xy % WIMG;
    const float v = pf[i];
    if (v >= EPS_F) {
      const double m = (double)(multv(y) * multv(x));
      c  += m;
      a1 += m * (double)v;
      a2 += m * (double)v * (double)v;
    }
  }
  sc[tid] = c; s1[tid] = a1; s2[tid] = a2;
  __syncthreads();
  for (int s = 128; s > 0; s >>= 1) {
    if (tid < s) {
      sc[tid] += sc[tid + s];
      s1[tid] += s1[tid + s];
      s2[tid] += s2[tid + s];
    }
    __syncthreads();
  }
  if (tid == 0) {
    const double m  = s1[0] / sc[0];
    const double m2 = s2[0] / sc[0];
    double var = m2 - m * m; if (var < 0.0) var = 0.0;
    thr_out[0] = (float)(2.0 * sqrt(var) / 5.0);   // NUM_CLASSES = 5
  }
}

// ---------------------------------------------------------------------------
// Pass 2: main STARFM kernel. One 256-thread block (8 wave32) per 16x16 output
// tile per (b,c). Stages the 30x30 input region (5 planes) into LDS via the
// gfx1250 async global->LDS path, then each lane computes one output pixel.
// Loop order dy -> dx -> t (t unrolled) so the pred_coarse LDS load and the
// spatial-distance polynomial are shared across the two time slices.
// ---------------------------------------------------------------------------
__global__ void __launch_bounds__(256)
starfm_main(const float* __restrict__ pc, const float* __restrict__ pf,
            const float* __restrict__ pd, const float* __restrict__ thrp,
            float* __restrict__ out) {
  // planes: 0,1 = prior_coarse t0,t1 ; 2,3 = prior_fine t0,t1 ; 4 = pred_coarse
  __shared__ float sm[5][REG][RSTRIDE];

  const int bc      = blockIdx.y;           // b*C_ + c, 0..7
  const int tileIdx = blockIdx.x;           // 0..35
  const int ty0 = (tileIdx / (OH_ / TILE)) * TILE;
  const int tx0 = (tileIdx % (OH_ / TILE)) * TILE;
  const int lt  = threadIdx.y * TILE + threadIdx.x;

  const float* base0 = pc + (size_t)(bc * T_ + 0) * PLANE;
  const float* base1 = pc + (size_t)(bc * T_ + 1) * PLANE;
  const float* base2 = pf + (size_t)(bc * T_ + 0) * PLANE;
  const float* base3 = pf + (size_t)(bc * T_ + 1) * PLANE;
  const float* base4 = pd + (size_t)bc * PLANE;

  // stage the 5 x 30x30 region into LDS (async DMA, ASYNCcnt-tracked)
  for (int i = lt; i < 5 * REG * REG; i += 256) {
    const int p   = i / (REG * REG);
    const int rem = i - p * (REG * REG);
    const int r   = rem / REG;
    const int cx  = rem - r * REG;
    const float* g =
        (p == 0 ? base0 : p == 1 ? base1 : p == 2 ? base2 : p == 3 ? base3 : base4)
        + (ty0 + r) * WIMG + (tx0 + cx);
    copy_f32_to_lds(g, &sm[p][r][cx]);
  }
  wait_async_lds();
  __syncthreads();

  const float thr = *thrp;
  const int ty = threadIdx.y, tx = threadIdx.x;

  // center values (window center = +7,+7)
  const float pcC0 = sm[0][ty + 7][tx + 7];
  const float pcC1 = sm[1][ty + 7][tx + 7];
  const float pfC0 = sm[2][ty + 7][tx + 7];
  const float pfC1 = sm[3][ty + 7][tx + 7];
  const float pdC  = sm[4][ty + 7][tx + 7];

  const float specLim = fmaxf(fabsf(pcC0 - pfC0), fabsf(pcC1 - pfC1)) + SPEC_U;
  const float tempLim = fmaxf(fabsf(pcC0 - pdC ), fabsf(pcC1 - pdC )) + TEMP_U;

  const bool cval0 = (pfC0 >= EPS_F);
  const bool cval1 = (pfC1 >= EPS_F);

  float S = 0.0f, P = 0.0f;
#pragma unroll 1
  for (int dy = 0; dy < WS_; ++dy) {
    const float fy  = (float)(dy - 7);
    const float fy2 = fy * fy;
    const float* rowc0 = &sm[0][ty + dy][tx];
    const float* rowc1 = &sm[1][ty + dy][tx];
    const float* rowf0 = &sm[2][ty + dy][tx];
    const float* rowf1 = &sm[3][ty + dy][tx];
    const float* rowd  = &sm[4][ty + dy][tx];
#pragma unroll
    for (int dx = 0; dx < WS_; ++dx) {
      const float fx   = (float)(dx - 7);
      const float spat = (fy2 + fx * fx) * (1.0f / 7.0f) + 1.0f;
      const float pdv  = rowd[dx];

      // ---- t = 0 ----
      {
        const float pcv = rowc0[dx];
        const float pfv = rowf0[dx];
        const float spec = fabsf(pcv - pfv);
        const float temp = fabsf(pcv - pdv);
        const float w =
            fast_rcp(__logf(spec + 1.0f) * __logf(temp + 1.0f) * spat + EPS_F);
        const bool keep = (spec < specLim) && (temp < tempLim) &&
                          (pfv >= EPS_F) && cval0 && (fabsf(pfv - pfC0) < thr);
        const float mw = keep ? w : 0.0f;
        S += mw;
        P += mw * (pdv + pfv - pcv);
      }
      // ---- t = 1 ----
      {
        const float pcv = rowc1[dx];
        const float pfv = rowf1[dx];
        const float spec = fabsf(pcv - pfv);
        const float temp = fabsf(pcv - pdv);
        const float w =
            fast_rcp(__logf(spec + 1.0f) * __logf(temp + 1.0f) * spat + EPS_F);
        const bool keep = (spec < specLim) && (temp < tempLim) &&
                          (pfv >= EPS_F) && cval1 && (fabsf(pfv - pfC1) < thr);
        const float mw = keep ? w : 0.0f;
        S += mw;
        P += mw * (pdv + pfv - pcv);
      }
    }
  }

  const int oy = ty0 + ty, ox = tx0 + tx;
  out[((size_t)bc * OH_ + oy) * OH_ + ox] = P / (S + EPS_F);
}

// ---------------------------------------------------------------------------
extern "C" void kernel_launch(void* const* d_in, const int* in_sizes, int n_in,
                              void* d_out, int out_size, void* d_ws, size_t ws_size,
                              hipStream_t stream) {
  const float* pc = (const float*)d_in[0];   // (2,4,2,110,110)
  const float* pf = (const float*)d_in[1];   // (2,4,2,110,110)
  const float* pd = (const float*)d_in[2];   // (2,4,110,110)
  float* out = (float*)d_out;                // (2,4,96,96)
  float* thr = (float*)d_ws;                 // 1 float of scratch

  starfm_stats<<<1, 256, 0, stream>>>(pf, thr);
  starfm_main<<<dim3((OH_ / TILE) * (OH_ / TILE), B_ * C_), dim3(TILE, TILE), 0, stream>>>(
      pc, pf, pd, thr, out);
}